// FpgnnModel_19258633356005
// MI455X (gfx1250) — compile-verified
//
#include <hip/hip_runtime.h>
#include <math.h>

// ---------------- model dims ----------------
#define BATCH   512
#define NATOM   128
#define NFEAT   133
#define NHID    60
#define NHEADS  8
#define HID     300
#define FP_DIM  1489
#define FP2     512
#define ALPHA   0.2f
#define NEGBIG  -9e15f

typedef float v2f __attribute__((ext_vector_type(2)));
typedef float v8f __attribute__((ext_vector_type(8)));

// D = A(16x4) * B(4x16) + C, fp32 WMMA (CDNA5)
__device__ __forceinline__ v8f wmma_k4(v2f a, v2f b, v8f c) {
  return __builtin_amdgcn_wmma_f32_16x16x4_f32(false, a, false, b, (short)0, c,
                                               false, false);
}

__device__ __forceinline__ float wave_max32(float v) {
#pragma unroll
  for (int o = 16; o > 0; o >>= 1) v = fmaxf(v, __shfl_xor(v, o, 32));
  return v;
}
__device__ __forceinline__ float wave_sum32(float v) {
#pragma unroll
  for (int o = 16; o > 0; o >>= 1) v += __shfl_xor(v, o, 32);
  return v;
}
__device__ __forceinline__ float elu1(float v)  { return v > 0.f ? v : (__expf(v) - 1.f); }
__device__ __forceinline__ float lrelu(float v) { return v > 0.f ? v : ALPHA * v; }

// =====================================================================
// Kernel 1: fused multi-head GAT layer. One workgroup per (b, head).
//   Wh = x[b] @ W_heads[h]  (WMMA, into LDS, 60 cols padded to 64)
//   s,d = Wh . a_src/a_dst
//   att = softmax(mask(leakyrelu(s_i + d_j)))   (per-wave row softmax)
//   h_cat[b,:,h*60:(h+1)*60] = elu(att @ Wh)    (WMMA)
// =====================================================================
__global__ __launch_bounds__(256) void gat_head_fused(
    const float* __restrict__ x, const int* __restrict__ adj,
    const float* __restrict__ W_heads, const float* __restrict__ a_src,
    const float* __restrict__ a_dst, float* __restrict__ h_cat) {
  __shared__ float Wl[136 * 64];       // W_heads[h], K padded 133->136, N 60->64
  __shared__ float Whl[128 * 64];      // Wh
  __shared__ float asl[64], adl[64];
  __shared__ float sv[128], dvv[128];
  __shared__ float attl[8][16 * 128];  // per-wave attention tile

  const int tid = threadIdx.x;
  const int lane = tid & 31;
  const int w = tid >> 5;
  const int hf = lane >> 4;   // lane half (K split for f32 WMMA frags)
  const int lr = lane & 15;
  const int b  = blockIdx.x >> 3;
  const int hh = blockIdx.x & 7;

  const float* Wg = W_heads + (size_t)hh * NFEAT * NHID;
  for (int idx = tid; idx < 136 * 64; idx += 256) {
    int k = idx >> 6, n = idx & 63;
    Wl[idx] = (k < NFEAT && n < NHID) ? Wg[k * NHID + n] : 0.f;
  }
  if (tid < 64) {
    asl[tid] = (tid < NHID) ? a_src[hh * NHID + tid] : 0.f;
    adl[tid] = (tid < NHID) ? a_dst[hh * NHID + tid] : 0.f;
  }
  __syncthreads();

  // ---- Wh GEMM: each wave does 16 rows x 64 cols ----
  const float* xg = x + (size_t)b * NATOM * NFEAT;
  {
    v8f acc[4] = {};
    const int row = w * 16 + lr;
    for (int k0 = 0; k0 < 136; k0 += 4) {
      int k = k0 + 2 * hf;
      v2f a;
      a.x = (k < NFEAT) ? xg[row * NFEAT + k] : 0.f;
      a.y = (k + 1 < NFEAT) ? xg[row * NFEAT + k + 1] : 0.f;
#pragma unroll
      for (int t = 0; t < 4; ++t) {
        v2f bb;
        bb.x = Wl[k * 64 + 16 * t + lr];
        bb.y = Wl[(k + 1) * 64 + 16 * t + lr];
        acc[t] = wmma_k4(a, bb, acc[t]);
      }
    }
#pragma unroll
    for (int t = 0; t < 4; ++t)
#pragma unroll
      for (int r = 0; r < 8; ++r)
        Whl[(w * 16 + r + 8 * hf) * 64 + 16 * t + lr] = acc[t][r];
  }
  __syncthreads();

  // ---- s, d vectors ----
  if (tid < 128) {
    float a = 0.f;
    for (int o = 0; o < 64; ++o) a += Whl[tid * 64 + o] * asl[o];
    sv[tid] = a;
  } else {
    int i = tid - 128;
    float a = 0.f;
    for (int o = 0; o < 64; ++o) a += Whl[i * 64 + o] * adl[o];
    dvv[i] = a;
  }
  __syncthreads();

  // ---- masked softmax rows (wave w owns rows 16w..16w+15) ----
  const int* adjb = adj + (size_t)b * NATOM * NATOM;
  const int i0 = w * 16;
  for (int ii = 0; ii < 16; ++ii) {
    const int i = i0 + ii;
    const float si = sv[i];
    float ev[4];
    float m = -INFINITY;
#pragma unroll
    for (int jj = 0; jj < 4; ++jj) {
      int j = lane + 32 * jj;
      float e = lrelu(si + dvv[j]);
      e = (adjb[i * NATOM + j] > 0) ? e : NEGBIG;
      ev[jj] = e;
      m = fmaxf(m, e);
    }
    m = wave_max32(m);
    float s = 0.f;
#pragma unroll
    for (int jj = 0; jj < 4; ++jj) { ev[jj] = __expf(ev[jj] - m); s += ev[jj]; }
    s = wave_sum32(s);
    float inv = 1.f / s;
#pragma unroll
    for (int jj = 0; jj < 4; ++jj) attl[w][ii * 128 + lane + 32 * jj] = ev[jj] * inv;
  }

  // ---- att @ Wh, elu, scatter into h_cat ----
  {
    v8f acc[4] = {};
    for (int k0 = 0; k0 < 128; k0 += 4) {
      int k = k0 + 2 * hf;
      v2f a;
      a.x = attl[w][lr * 128 + k];
      a.y = attl[w][lr * 128 + k + 1];
#pragma unroll
      for (int t = 0; t < 4; ++t) {
        v2f bb;
        bb.x = Whl[k * 64 + 16 * t + lr];
        bb.y = Whl[(k + 1) * 64 + 16 * t + lr];
        acc[t] = wmma_k4(a, bb, acc[t]);
      }
    }
    float* hb = h_cat + (size_t)b * NATOM * (NHEADS * NHID) + hh * NHID;
#pragma unroll
    for (int t = 0; t < 4; ++t) {
      int col = 16 * t + lr;
      if (col < NHID) {
#pragma unroll
        for (int r = 0; r < 8; ++r) {
          int row = i0 + r + 8 * hf;
          hb[(size_t)row * (NHEADS * NHID) + col] = elu1(acc[t][r]);
        }
      }
    }
  }
}

// =====================================================================
// Generic fp32 WMMA GEMM: C[M,N] = act(A[M,K] @ B[K,N] + bias)
// grid = (M/128, ceil(N/64)), 256 threads; wave = 16 rows x 64 cols.
// =====================================================================
__global__ __launch_bounds__(256) void gemm_wmma_f32(
    const float* __restrict__ A, const float* __restrict__ Bm,
    const float* __restrict__ bias, float* __restrict__ C,
    int M, int N, int K, int lda, int ldb, int ldc, int act) {
  const int tid = threadIdx.x;
  const int lane = tid & 31;
  const int w = tid >> 5;
  const int hf = lane >> 4;
  const int lr = lane & 15;
  const int r0 = blockIdx.x * 128 + w * 16;
  const int cb = blockIdx.y * 64;

  v8f acc[4] = {};
  const float* Arow = A + (size_t)(r0 + lr) * lda;
  for (int k0 = 0; k0 < K; k0 += 4) {
    int k = k0 + 2 * hf;
    v2f a;
    a.x = (k < K) ? Arow[k] : 0.f;
    a.y = (k + 1 < K) ? Arow[k + 1] : 0.f;
#pragma unroll
    for (int t = 0; t < 4; ++t) {
      int col = cb + 16 * t + lr;
      bool cv = col < N;
      v2f bb;
      bb.x = (cv && k < K) ? Bm[(size_t)k * ldb + col] : 0.f;
      bb.y = (cv && k + 1 < K) ? Bm[(size_t)(k + 1) * ldb + col] : 0.f;
      acc[t] = wmma_k4(a, bb, acc[t]);
    }
  }
#pragma unroll
  for (int t = 0; t < 4; ++t) {
    int col = cb + 16 * t + lr;
    if (col < N) {
      float bv = bias ? bias[col] : 0.f;
#pragma unroll
      for (int r = 0; r < 8; ++r) {
        int row = r0 + r + 8 * hf;
        if (row < M) {
          float v = acc[t][r] + bv;
          if (act == 1) v = fmaxf(v, 0.f);
          C[(size_t)row * ldc + col] = v;
        }
      }
    }
  }
}

// =====================================================================
// so/do = Wo . ao_src / ao_dst  (one wave per row of 300)
// =====================================================================
__global__ __launch_bounds__(256) void rowdots(
    const float* __restrict__ Wo, const float* __restrict__ aos,
    const float* __restrict__ aod, float* __restrict__ so,
    float* __restrict__ dvo) {
  const int lane = threadIdx.x & 31;
  const int w = threadIdx.x >> 5;
  const int row = blockIdx.x * 8 + w;
  const float* r = Wo + (size_t)row * HID;
  float s1 = 0.f, s2 = 0.f;
  for (int o = lane; o < HID; o += 32) {
    float v = r[o];
    s1 += v * aos[o];
    s2 += v * aod[o];
  }
  s1 = wave_sum32(s1);
  s2 = wave_sum32(s2);
  if (lane == 0) { so[row] = s1; dvo[row] = s2; }
}

// =====================================================================
// Output GAT attention: per b, hoRaw = softmax(mask(lrelu(so_i+do_j))) @ Wo[b]
// =====================================================================
__global__ __launch_bounds__(256) void gat_out_attn(
    const float* __restrict__ Wo, const float* __restrict__ so,
    const float* __restrict__ dvo, const int* __restrict__ adj,
    float* __restrict__ hoRaw) {
  __shared__ float sl[128], dl[128];
  __shared__ float attl[8][16 * 128];
  const int tid = threadIdx.x;
  const int lane = tid & 31;
  const int w = tid >> 5;
  const int hf = lane >> 4;
  const int lr = lane & 15;
  const int b = blockIdx.x;
  if (tid < 128) sl[tid] = so[b * NATOM + tid];
  else           dl[tid - 128] = dvo[b * NATOM + tid - 128];
  __syncthreads();

  const int* adjb = adj + (size_t)b * NATOM * NATOM;
  const int i0 = w * 16;
  for (int ii = 0; ii < 16; ++ii) {
    const int i = i0 + ii;
    const float si = sl[i];
    float ev[4];
    float m = -INFINITY;
#pragma unroll
    for (int jj = 0; jj < 4; ++jj) {
      int j = lane + 32 * jj;
      float e = lrelu(si + dl[j]);
      e = (adjb[i * NATOM + j] > 0) ? e : NEGBIG;
      ev[jj] = e;
      m = fmaxf(m, e);
    }
    m = wave_max32(m);
    float s = 0.f;
#pragma unroll
    for (int jj = 0; jj < 4; ++jj) { ev[jj] = __expf(ev[jj] - m); s += ev[jj]; }
    s = wave_sum32(s);
    float inv = 1.f / s;
#pragma unroll
    for (int jj = 0; jj < 4; ++jj) attl[w][ii * 128 + lane + 32 * jj] = ev[jj] * inv;
  }

  const float* Wob = Wo + (size_t)b * NATOM * HID;
  float* hob = hoRaw + (size_t)b * NATOM * HID;
  for (int cb = 0; cb < 320; cb += 64) {   // 300 cols in chunks of 64
    v8f acc[4] = {};
    for (int k0 = 0; k0 < 128; k0 += 4) {
      int k = k0 + 2 * hf;
      v2f a;
      a.x = attl[w][lr * 128 + k];
      a.y = attl[w][lr * 128 + k + 1];
#pragma unroll
      for (int t = 0; t < 4; ++t) {
        int col = cb + 16 * t + lr;
        bool cv = col < HID;
        v2f bb;
        bb.x = cv ? Wob[(size_t)k * HID + col] : 0.f;
        bb.y = cv ? Wob[(size_t)(k + 1) * HID + col] : 0.f;
        acc[t] = wmma_k4(a, bb, acc[t]);
      }
    }
#pragma unroll
    for (int t = 0; t < 4; ++t) {
      int col = cb + 16 * t + lr;
      if (col < HID) {
#pragma unroll
        for (int r = 0; r < 8; ++r) {
          int row = i0 + r + 8 * hf;
          hob[(size_t)row * HID + col] = acc[t][r];
        }
      }
    }
  }
}

// =====================================================================
// gat[b,:] = mean_n log_softmax(elu(hoRaw[b,n,:]))  (over 300 features)
// =====================================================================
__global__ __launch_bounds__(256) void gat_reduce(
    const float* __restrict__ hoRaw, float* __restrict__ gat) {
  __shared__ float facc[304];
  __shared__ float wred[8];
  const int tid = threadIdx.x;
  const int lane = tid & 31;
  const int w = tid >> 5;
  const int b = blockIdx.x;
  for (int i = tid; i < 304; i += 256) facc[i] = 0.f;
  __syncthreads();
  const float* base = hoRaw + (size_t)b * NATOM * HID;
  const bool has2 = (tid + 256) < HID;
  for (int n = 0; n < NATOM; ++n) {
    const float* row = base + (size_t)n * HID;
    float v0 = elu1(row[tid]);
    float v1 = has2 ? elu1(row[tid + 256]) : -INFINITY;
    float m = fmaxf(v0, v1);
    m = wave_max32(m);
    if (lane == 0) wred[w] = m;
    __syncthreads();
    float M = wred[0];
#pragma unroll
    for (int k = 1; k < 8; ++k) M = fmaxf(M, wred[k]);
    __syncthreads();
    float s = __expf(v0 - M) + (has2 ? __expf(v1 - M) : 0.f);
    s = wave_sum32(s);
    if (lane == 0) wred[w] = s;
    __syncthreads();
    float S = 0.f;
#pragma unroll
    for (int k = 0; k < 8; ++k) S += wred[k];
    float L = __logf(S);
    facc[tid] += v0 - M - L;
    if (has2) facc[tid + 256] += v1 - M - L;
    __syncthreads();
  }
  const float invN = 1.f / (float)NATOM;
  gat[(size_t)b * HID + tid] = facc[tid] * invN;          // tid < 256 < 300
  if (has2) gat[(size_t)b * HID + tid + 256] = facc[tid + 256] * invN;
}

// =====================================================================
// out = sigmoid(z2 @ ffn2_w + ffn2_b)   (TASK == 1)
// =====================================================================
__global__ __launch_bounds__(256) void ffn2_sig(
    const float* __restrict__ z2, const float* __restrict__ w2,
    const float* __restrict__ b2, float* __restrict__ out) {
  const int lane = threadIdx.x & 31;
  const int w = threadIdx.x >> 5;
  const int row = blockIdx.x * 8 + w;
  float s = 0.f;
  for (int o = lane; o < HID; o += 32) s += z2[(size_t)row * HID + o] * w2[o];
  s = wave_sum32(s);
  if (lane == 0) out[row] = 1.f / (1.f + __expf(-(s + b2[0])));
}

// =====================================================================
extern "C" void kernel_launch(void* const* d_in, const int* in_sizes, int n_in,
                              void* d_out, int out_size, void* d_ws,
                              size_t ws_size, hipStream_t stream) {
  (void)in_sizes; (void)n_in; (void)out_size; (void)ws_size;
  const float* x       = (const float*)d_in[0];
  const int*   adj     = (const int*)d_in[1];
  const float* fpv     = (const float*)d_in[2];
  const float* W_heads = (const float*)d_in[3];
  const float* a_src   = (const float*)d_in[4];
  const float* a_dst   = (const float*)d_in[5];
  const float* W_out   = (const float*)d_in[6];
  const float* ao_src  = (const float*)d_in[7];
  const float* ao_dst  = (const float*)d_in[8];
  const float* fc1_w   = (const float*)d_in[9];
  const float* fc1_b   = (const float*)d_in[10];
  const float* fc2_w   = (const float*)d_in[11];
  const float* fc2_b   = (const float*)d_in[12];
  const float* fcg_w   = (const float*)d_in[13];
  const float* fcg_b   = (const float*)d_in[14];
  const float* fcf_w   = (const float*)d_in[15];
  const float* fcf_b   = (const float*)d_in[16];
  const float* ffn1_w  = (const float*)d_in[17];
  const float* ffn1_b  = (const float*)d_in[18];
  const float* ffn2_w  = (const float*)d_in[19];
  const float* ffn2_b  = (const float*)d_in[20];

  float* ws = (float*)d_ws;
  size_t off = 0;
  float* h_cat = ws + off; off += (size_t)BATCH * NATOM * NHEADS * NHID; // 31.4M
  float* Wo    = ws + off; off += (size_t)BATCH * NATOM * HID;           // 19.7M
  float* so    = ws + off; off += (size_t)BATCH * NATOM;
  float* dvo   = ws + off; off += (size_t)BATCH * NATOM;
  float* gat   = ws + off; off += (size_t)BATCH * HID;
  float* f1    = ws + off; off += (size_t)BATCH * FP2;
  float* f2    = ws + off; off += (size_t)BATCH * HID;
  float* zb    = ws + off; off += (size_t)BATCH * 2 * HID;
  float* z2    = ws + off; off += (size_t)BATCH * HID;
  float* hoRaw = h_cat;  // h_cat is dead after the Wo GEMM; reuse for ho

  // 1. fused first GAT layer
  gat_head_fused<<<BATCH * NHEADS, 256, 0, stream>>>(x, adj, W_heads, a_src,
                                                     a_dst, h_cat);
  // 2. Wo = h_cat @ W_out   [65536x480]·[480x300]
  gemm_wmma_f32<<<dim3(BATCH * NATOM / 128, 5), 256, 0, stream>>>(
      h_cat, W_out, nullptr, Wo, BATCH * NATOM, HID, NHEADS * NHID,
      NHEADS * NHID, HID, HID, 0);
  // 3. per-row attention scalars
  rowdots<<<BATCH * NATOM / 8, 256, 0, stream>>>(Wo, ao_src, ao_dst, so, dvo);
  // 4. output attention layer (writes into reused h_cat region)
  gat_out_attn<<<BATCH, 256, 0, stream>>>(Wo, so, dvo, adj, hoRaw);
  // 5. elu + log_softmax + mean over atoms
  gat_reduce<<<BATCH, 256, 0, stream>>>(hoRaw, gat);
  // 6. FPN branch
  gemm_wmma_f32<<<dim3(BATCH / 128, FP2 / 64), 256, 0, stream>>>(
      fpv, fc1_w, fc1_b, f1, BATCH, FP2, FP_DIM, FP_DIM, FP2, FP2, 1);
  gemm_wmma_f32<<<dim3(BATCH / 128, 5), 256, 0, stream>>>(
      f1, fc2_w, fc2_b, f2, BATCH, HID, FP2, FP2, HID, HID, 0);
  // 7. fusion: g -> zb[:, :300], f -> zb[:, 300:600]
  gemm_wmma_f32<<<dim3(BATCH / 128, 5), 256, 0, stream>>>(
      gat, fcg_w, fcg_b, zb, BATCH, HID, HID, HID, HID, 2 * HID, 1);
  gemm_wmma_f32<<<dim3(BATCH / 128, 5), 256, 0, stream>>>(
      f2, fcf_w, fcf_b, zb + HID, BATCH, HID, HID, HID, HID, 2 * HID, 1);
  gemm_wmma_f32<<<dim3(BATCH / 128, 5), 256, 0, stream>>>(
      zb, ffn1_w, ffn1_b, z2, BATCH, HID, 2 * HID, 2 * HID, HID, HID, 1);
  // 8. final linear + sigmoid
  ffn2_sig<<<BATCH / 8, 256, 0, stream>>>(z2, ffn2_w, ffn2_b, (float*)d_out);
}